// ADAGCN_LP_22995254903260
// MI455X (gfx1250) — compile-verified
//
#include <hip/hip_runtime.h>
#include <hip/hip_bf16.h>

// ---------------------------------------------------------------------------
// Problem constants (match reference)
// ---------------------------------------------------------------------------
#define GN 100000      // nodes per graph
#define GE 1600000     // edges per graph
#define GD 128         // feature dim
#define GB 100000      // pairs per domain
#define ROWTILES (GN / 16)   // 6250 (N divisible by 16)

typedef float v2f __attribute__((ext_vector_type(2)));
typedef float v8f __attribute__((ext_vector_type(8)));

// ---------------------------------------------------------------------------
// Utility kernels
// ---------------------------------------------------------------------------
__global__ void k_fill(float* __restrict__ p, float v, long long n) {
    long long g = (long long)blockIdx.x * blockDim.x + threadIdx.x;
    if (g < n) p[g] = v;
}

__global__ void k_deg(const int* __restrict__ dst, float* __restrict__ deg, int ne) {
    int g = blockIdx.x * blockDim.x + threadIdx.x;
    if (g < ne) atomicAdd(deg + dst[g], 1.0f);
}

__global__ void k_rsqrt(float* __restrict__ p, int n) {
    int g = blockIdx.x * blockDim.x + threadIdx.x;
    if (g < n) p[g] = rsqrtf(p[g]);
}

// ---------------------------------------------------------------------------
// Dense GEMM  Y[N x 128] = X[N x 128] @ W[128 x 128]  via V_WMMA_F32_16X16X4_F32
// Block = 256 threads = 8 waves. W staged in LDS (64KB). Each wave owns one
// 16-row strip and computes all 8 column tiles (16 cols each), K in steps of 4.
//
// ISA layouts (CDNA5 ISA §7.12.2):
//   A (16x4 f32):  lanes 0-15 -> M=lane, v0=K0 v1=K1 ; lanes 16-31 -> K2,K3
//   B (4x16 f32):  vgpr v: lanes 0-15 -> K=v, N=lane ; lanes 16-31 -> K=v+2
//   C/D (16x16):   vgpr r: lanes 0-15 -> M=r, N=lane ; lanes 16-31 -> M=r+8
// ---------------------------------------------------------------------------
__global__ void __launch_bounds__(256)
k_gemm_wmma(const float* __restrict__ X, const float* __restrict__ Wg,
            float* __restrict__ Y, int nRowTiles) {
    __shared__ float Wl[GD * GD];   // 64 KB (well under CDNA5's 320KB/WGP)
    {
        const float4* s = (const float4*)Wg;
        float4*       d = (float4*)Wl;
        for (int i = threadIdx.x; i < GD * GD / 4; i += 256) d[i] = s[i];
    }
    __syncthreads();

    const int wave = threadIdx.x >> 5;
    const int lane = threadIdx.x & 31;
    const int rowTile = blockIdx.x * 8 + wave;
    if (rowTile >= nRowTiles) return;           // uniform per wave: EXEC stays all-1s

    const int row0 = rowTile * 16;
    const int half = lane >> 4;                  // 0: K pair {0,1}, 1: {2,3}
    const int m    = lane & 15;

    v8f acc[8] = {};
    for (int kk = 0; kk < 32; ++kk) {
        const int k0 = kk * 4 + half * 2;
        v2f a;
        const float* ap = X + (size_t)(row0 + m) * GD + k0;
        a.x = ap[0];
        a.y = ap[1];
#pragma unroll
        for (int t = 0; t < 8; ++t) {
            v2f b;
            b.x = Wl[(k0 + 0) * GD + t * 16 + m];
            b.y = Wl[(k0 + 1) * GD + t * 16 + m];
            acc[t] = __builtin_amdgcn_wmma_f32_16x16x4_f32(
                false, a, false, b, (short)0, acc[t], false, false);
        }
    }
#pragma unroll
    for (int t = 0; t < 8; ++t) {
#pragma unroll
        for (int r = 0; r < 8; ++r) {
            Y[(size_t)(row0 + r + 8 * half) * GD + t * 16 + m] = acc[t][r];
        }
    }
}

// ---------------------------------------------------------------------------
// Edge scatter: AGG[dst] += H[src] * dinv[src]*dinv[dst]
// 32 lanes per edge, float4 per lane (coalesced row gather), f32 atomics.
// ---------------------------------------------------------------------------
__global__ void k_scatter(const float* __restrict__ H, const int* __restrict__ src,
                          const int* __restrict__ dst, const float* __restrict__ dinv,
                          float* __restrict__ AGG, long long nWork) {
    long long g = (long long)blockIdx.x * blockDim.x + threadIdx.x;
    if (g >= nWork) return;
    const int e = (int)(g >> 5);
    const int c = (int)(g & 31);
    const int s = src[e];
    const int d = dst[e];
    const float norm = dinv[s] * dinv[d];
    const float4 hv = ((const float4*)(H + (size_t)s * GD))[c];
    float* out = AGG + (size_t)d * GD + c * 4;
    atomicAdd(out + 0, hv.x * norm);
    atomicAdd(out + 1, hv.y * norm);
    atomicAdd(out + 2, hv.z * norm);
    atomicAdd(out + 3, hv.w * norm);
}

// AGG[i,d] = (AGG[i,d] + H[i,d]*dinv[i]^2 + bias[d])  (optional relu), in place
__global__ void k_finalize(float* __restrict__ AGG, const float* __restrict__ H,
                           const float* __restrict__ dinv, const float* __restrict__ bias,
                           int doRelu, long long total) {
    long long g = (long long)blockIdx.x * blockDim.x + threadIdx.x;
    if (g >= total) return;
    const int n = (int)(g >> 7);
    const int d = (int)(g & 127);
    const float di = dinv[n];
    float v = AGG[g] + H[g] * di * di + bias[d];
    if (doRelu) v = fmaxf(v, 0.0f);
    AGG[g] = v;
}

// ---------------------------------------------------------------------------
// Pair loss partials: one wave per (user,item) pair
// ---------------------------------------------------------------------------
__device__ __forceinline__ float wave_reduce(float v) {
#pragma unroll
    for (int off = 16; off > 0; off >>= 1) v += __shfl_xor(v, off, 32);
    return v;
}

__global__ void __launch_bounds__(256)
k_pairs(const float* __restrict__ XO,
        const int* __restrict__ user, const int* __restrict__ item,
        const int* __restrict__ labels, const int* __restrict__ numUserP,
        const float* __restrict__ clsW, const float* __restrict__ clsB,
        const float* __restrict__ duW, const float* __restrict__ duB,
        const float* __restrict__ diW, const float* __restrict__ diB,
        float* __restrict__ accum, int base, int Bn) {
    const int wave = threadIdx.x >> 5;
    const int lane = threadIdx.x & 31;
    const int p = blockIdx.x * 8 + wave;
    if (p >= Bn) return;

    const int u  = user[p];
    const int it = item[p] + numUserP[0];

    const float4 uf  = ((const float4*)(XO + (size_t)u * GD))[lane];
    const float4 vf  = ((const float4*)(XO + (size_t)it * GD))[lane];
    const float4 cwU = ((const float4*)clsW)[lane];
    const float4 cwI = ((const float4*)clsW)[lane + 32];
    const float4 dw  = ((const float4*)duW)[lane];
    const float4 iw  = ((const float4*)diW)[lane];

    float zp  = uf.x * cwU.x + uf.y * cwU.y + uf.z * cwU.z + uf.w * cwU.w
              + vf.x * cwI.x + vf.y * cwI.y + vf.z * cwI.z + vf.w * cwI.w;
    float dup = uf.x * dw.x + uf.y * dw.y + uf.z * dw.z + uf.w * dw.w;
    float dip = vf.x * iw.x + vf.y * iw.y + vf.z * iw.z + vf.w * iw.w;

    zp  = wave_reduce(zp);
    dup = wave_reduce(dup);
    dip = wave_reduce(dip);

    if (lane == 0) {
        const float z = zp + clsB[0];
        const float y = (float)labels[p];
        // numerically stable softplus
        const float common = log1pf(expf(-fabsf(z)));
        const float sp_pos = common + fmaxf(z, 0.0f);    // softplus(z)
        const float sp_neg = common + fmaxf(-z, 0.0f);   // softplus(-z)
        const float bce = y * sp_neg + (1.0f - y) * sp_pos;
        const float cu = 1.0f / (1.0f + expf(-(dup + duB[0])));
        const float ci = 1.0f / (1.0f + expf(-(dip + diB[0])));
        atomicAdd(accum + base + 0, bce);
        atomicAdd(accum + base + 1, cu);
        atomicAdd(accum + base + 2, ci);
    }
}

__global__ void k_final(const float* __restrict__ accum, float* __restrict__ out, int Bn) {
    const float invB = 1.0f / (float)Bn;
    const float clf = accum[0] * invB + accum[3] * invB;
    const float dom = fabsf(accum[1] * invB - accum[4] * invB)
                    + fabsf(accum[2] * invB - accum[5] * invB);
    out[0] = clf + dom;
}

// ---------------------------------------------------------------------------
// Host-side per-graph pipeline (all launches on `stream`, graph-capturable)
// ---------------------------------------------------------------------------
static void run_graph(const float* feats, const int* esrc, const int* edst,
                      const float* W1, const float* b1,
                      const float* W2, const float* b2,
                      float* bufA, float* bufB, float* dinv, hipStream_t stream) {
    const long long ND = (long long)GN * GD;
    const long long nWork = (long long)GE * 32;
    const int gemmBlocks = (ROWTILES + 7) / 8;
    const int ndBlocks = (int)((ND + 255) / 256);
    const int wkBlocks = (int)((nWork + 255) / 256);

    // degree (self loop => init 1.0) -> dinv = rsqrt(deg)
    k_fill<<<(GN + 255) / 256, 256, 0, stream>>>(dinv, 1.0f, (long long)GN);
    k_deg<<<(GE + 255) / 256, 256, 0, stream>>>(edst, dinv, GE);
    k_rsqrt<<<(GN + 255) / 256, 256, 0, stream>>>(dinv, GN);

    // layer 1: bufA = x@W1 ; bufB = relu(scatter + self + b1)
    k_gemm_wmma<<<gemmBlocks, 256, 0, stream>>>(feats, W1, bufA, ROWTILES);
    k_fill<<<ndBlocks, 256, 0, stream>>>(bufB, 0.0f, ND);
    k_scatter<<<wkBlocks, 256, 0, stream>>>(bufA, esrc, edst, dinv, bufB, nWork);
    k_finalize<<<ndBlocks, 256, 0, stream>>>(bufB, bufA, dinv, b1, 1, ND);

    // layer 2: bufA = h@W2 ; bufB = scatter + self + b2 (no relu)
    k_gemm_wmma<<<gemmBlocks, 256, 0, stream>>>(bufB, W2, bufA, ROWTILES);
    k_fill<<<ndBlocks, 256, 0, stream>>>(bufB, 0.0f, ND);
    k_scatter<<<wkBlocks, 256, 0, stream>>>(bufA, esrc, edst, dinv, bufB, nWork);
    k_finalize<<<ndBlocks, 256, 0, stream>>>(bufB, bufA, dinv, b2, 0, ND);
}

extern "C" void kernel_launch(void* const* d_in, const int* in_sizes, int n_in,
                              void* d_out, int out_size, void* d_ws, size_t ws_size,
                              hipStream_t stream) {
    (void)in_sizes; (void)n_in; (void)out_size; (void)ws_size;

    const float* feats_s = (const float*)d_in[0];
    const float* feats_t = (const float*)d_in[1];
    const float* W1      = (const float*)d_in[2];
    const float* b1      = (const float*)d_in[3];
    const float* W2      = (const float*)d_in[4];
    const float* b2      = (const float*)d_in[5];
    const float* clsW    = (const float*)d_in[6];
    const float* clsB    = (const float*)d_in[7];
    const float* duW     = (const float*)d_in[8];
    const float* duB     = (const float*)d_in[9];
    const float* diW     = (const float*)d_in[10];
    const float* diB     = (const float*)d_in[11];
    const int* esrc_s    = (const int*)d_in[12];
    const int* edst_s    = (const int*)d_in[13];
    const int* esrc_t    = (const int*)d_in[14];
    const int* edst_t    = (const int*)d_in[15];
    const int* user_s    = (const int*)d_in[16];
    const int* item_s    = (const int*)d_in[17];
    const int* labels_s  = (const int*)d_in[18];
    const int* user_t    = (const int*)d_in[19];
    const int* item_t    = (const int*)d_in[20];
    const int* labels_t  = (const int*)d_in[21];
    const int* numU_s    = (const int*)d_in[22];
    const int* numU_t    = (const int*)d_in[23];

    // workspace layout: bufA | bufB | dinv | accum   (~102.8 MB)
    const size_t NB = (size_t)GN * GD * sizeof(float);     // 51.2 MB, 256B aligned
    char* ws = (char*)d_ws;
    float* bufA  = (float*)ws;
    float* bufB  = (float*)(ws + NB);
    float* dinv  = (float*)(ws + 2 * NB);
    float* accum = (float*)(ws + 2 * NB + ((((size_t)GN * sizeof(float)) + 255) & ~(size_t)255));

    // zero loss accumulators [bce_s, cu_s, ci_s, bce_t, cu_t, ci_t]
    k_fill<<<1, 32, 0, stream>>>(accum, 0.0f, 8);

    // source domain
    run_graph(feats_s, esrc_s, edst_s, W1, b1, W2, b2, bufA, bufB, dinv, stream);
    k_pairs<<<(GB + 7) / 8, 256, 0, stream>>>(bufB, user_s, item_s, labels_s, numU_s,
                                              clsW, clsB, duW, duB, diW, diB,
                                              accum, 0, GB);
    // target domain (reuses buffers)
    run_graph(feats_t, esrc_t, edst_t, W1, b1, W2, b2, bufA, bufB, dinv, stream);
    k_pairs<<<(GB + 7) / 8, 256, 0, stream>>>(bufB, user_t, item_t, labels_t, numU_t,
                                              clsW, clsB, duW, duB, diW, diB,
                                              accum, 3, GB);

    k_final<<<1, 1, 0, stream>>>(accum, (float*)d_out, GB);
}